// NPTransitionPrior_10118942949543
// MI455X (gfx1250) — compile-verified
//
#include <hip/hip_runtime.h>
#include <math.h>

// Problem constants (match reference)
#define SLOPE  0.2f
#define NLAT   64
#define NB     32
#define LEN    128
#define TROW   130      // LAGS + LEN
#define NTOT   4096     // NB * LEN
#define D1     129      // LAGS*LATENT + 1
#define HID    64
#define SSTR   66       // padded LDS row stride (floats) to avoid bank conflicts
#define RESID_ELEMS (NB*LEN*NLAT)   // 262144

typedef __attribute__((ext_vector_type(2))) float v2f;
typedef __attribute__((ext_vector_type(8))) float v8f;

// Fused dual 16x64 f32 GEMM: both A operands (forward chain from sH, derivative
// chain from sD) share every B-fragment load of the weight matrix.
// B[k][col] = Wg[col*64 + k]  (W1[l]/W2[l] are [g][h] row-major and
// z[n,g] = sum_h A[n,h]*W[g,h], so B-fragment reads are contiguous in k).
// f32 WMMA operand layout: A lane = M (lane&15), VGPR pair = K, K+2 for lanes 16-31.
// Fully unrolled so all address arithmetic folds to compile-time offsets.
__device__ __forceinline__ void gemm16x64_dual(const float* Ah, const float* Ad,
                                               const float* __restrict__ Wg,
                                               int m, int kh, v8f zc[4], v8f uc[4])
{
    #pragma unroll
    for (int k0 = 0; k0 < HID; k0 += 4) {
        const int krow = k0 + 2 * kh;
        v2f ah = *(const v2f*)(Ah + m * SSTR + krow);   // 8B aligned: m*66+k, both even
        v2f ad = *(const v2f*)(Ad + m * SSTR + krow);
        #pragma unroll
        for (int nt = 0; nt < 4; ++nt) {
            const int col = nt * 16 + m;
            v2f bf = *(const v2f*)(Wg + (size_t)col * HID + krow); // 8B aligned
            zc[nt] = __builtin_amdgcn_wmma_f32_16x16x4_f32(
                false, ah, false, bf, (short)0, zc[nt], false, false);
            uc[nt] = __builtin_amdgcn_wmma_f32_16x16x4_f32(
                false, ad, false, bf, (short)0, uc[nt], false, false);
        }
    }
}

__global__ __launch_bounds__(128) void np_prior_main(
    const float* __restrict__ x,    // (32,130,64)
    const float* __restrict__ W0,   // (64,64,129)
    const float* __restrict__ b0,   // (64,64)
    const float* __restrict__ W1,   // (64,64,64)
    const float* __restrict__ b1,
    const float* __restrict__ W2,
    const float* __restrict__ b2,
    const float* __restrict__ Wout, // (64,64)
    const float* __restrict__ bout, // (64,)
    float* __restrict__ resid,      // (4096,64) = residuals, layout n*64+l
    float* __restrict__ wslog)      // (64,4096) log|dJ| per (l,n)
{
    __shared__ float sH[4][16][SSTR];   // per-wave forward activations (16 x 64)
    __shared__ float sD[4][16][SSTR];   // per-wave derivative chain

    const int l    = blockIdx.y;            // latent
    const int w    = threadIdx.x >> 5;      // wave in block (0..3)
    const int lane = threadIdx.x & 31;
    const int m    = lane & 15;             // M index for A-frag / N index for C-frag
    const int kh   = lane >> 4;             // wave half (K offset for A/B frags)

    const int row0 = blockIdx.x * 64 + w * 16;  // global row base n for this wave
    const int bb   = row0 >> 7;                 // batch
    const int t0   = row0 & 127;                // time base within batch

    // ---------------- layer 1: z1 = lags @ W0_lags^T ----------------
    // Fully unrolled: 32 K-steps x 4 col-tiles = 128 static WMMAs, all
    // address arithmetic folded into instruction offsets.
    v8f acc[4] = {};
    const float* xrow = x + ((size_t)(bb * TROW + t0 + m)) * NLAT; // row for lag 0
    const float* w0row = W0 + (size_t)(l * NLAT + m) * D1;         // col base (nt adds 16*D1)
    #pragma unroll
    for (int k0 = 0; k0 < 2 * NLAT; k0 += 4) {
        const int lag = k0 >> 6;
        const int j   = (k0 & 63) + 2 * kh;
        // lags[n, lag*64+j] = x[bb, t+lag, j]; two consecutive j -> float2 (8B aligned, j even)
        const float* pa = xrow + (size_t)lag * NLAT + j;
        v2f a; a.x = pa[0]; a.y = pa[1];
        const int krow = k0 + 2 * kh;
        #pragma unroll
        for (int nt = 0; nt < 4; ++nt) {
            // B[k][h] = W0[l,h,k]; row length 129 (odd) -> two scalar loads
            const float* pb = w0row + (size_t)(nt * 16) * D1 + krow;
            v2f bf; bf.x = pb[0]; bf.y = pb[1];
            acc[nt] = __builtin_amdgcn_wmma_f32_16x16x4_f32(
                false, a, false, bf, (short)0, acc[nt], false, false);
        }
    }

    // elementwise: z1 += x_t[:,l]*w0_last + b0 ; h1 = lrelu(z1); d1 = lrelu'(z1)*w0_last
    // C/D layout: VGPR v, lanes 0-15 -> (M=v, N=lane), lanes 16-31 -> (M=v+8, N=lane-16)
    #pragma unroll
    for (int nt = 0; nt < 4; ++nt) {
        const int h  = nt * 16 + m;
        const float w0l = W0[((size_t)(l * NLAT + h)) * D1 + (D1 - 1)];
        const float bv  = b0[l * NLAT + h];
        #pragma unroll
        for (int v = 0; v < 8; ++v) {
            const int mr = v + 8 * kh;
            const float xt = x[((size_t)(bb * TROW + t0 + mr + 2)) * NLAT + l];
            const float z  = acc[nt][v] + xt * w0l + bv;
            sH[w][mr][h] = (z >= 0.f) ? z : SLOPE * z;
            sD[w][mr][h] = ((z >= 0.f) ? 1.f : SLOPE) * w0l;
        }
    }
    __syncthreads();

    // ---------------- layer 2 (forward + derivative fused, shared B loads) ----
    v8f zc[4] = {}, uc[4] = {};
    gemm16x64_dual(&sH[w][0][0], &sD[w][0][0], W1 + (size_t)l * HID * HID, m, kh, zc, uc);

    #pragma unroll
    for (int nt = 0; nt < 4; ++nt) {
        const int g  = nt * 16 + m;
        const float bv = b1[l * NLAT + g];
        #pragma unroll
        for (int v = 0; v < 8; ++v) {
            const int mr = v + 8 * kh;
            const float z = zc[nt][v] + bv;
            sH[w][mr][g] = (z >= 0.f) ? z : SLOPE * z;
            sD[w][mr][g] = ((z >= 0.f) ? 1.f : SLOPE) * uc[nt][v];
        }
    }
    __syncthreads();

    // ---------------- layer 3 (fused) ----------------
    v8f zc2[4] = {}, uc2[4] = {};
    gemm16x64_dual(&sH[w][0][0], &sD[w][0][0], W2 + (size_t)l * HID * HID, m, kh, zc2, uc2);

    // elementwise + final dots with Wout (kept in registers, no LDS round-trip)
    float pr[8], pj[8];
    #pragma unroll
    for (int v = 0; v < 8; ++v) { pr[v] = 0.f; pj[v] = 0.f; }
    #pragma unroll
    for (int nt = 0; nt < 4; ++nt) {
        const int h  = nt * 16 + m;
        const float bv = b2[l * NLAT + h];
        const float wo = Wout[l * NLAT + h];
        #pragma unroll
        for (int v = 0; v < 8; ++v) {
            const float z  = zc2[nt][v] + bv;
            const float h3 = (z >= 0.f) ? z : SLOPE * z;
            const float d3 = ((z >= 0.f) ? 1.f : SLOPE) * uc2[nt][v];
            pr[v] += h3 * wo;
            pj[v] += d3 * wo;
        }
    }

    // reduce over the 16 lanes of each half-wave (h-dimension), then write
    #pragma unroll
    for (int v = 0; v < 8; ++v) {
        float r = pr[v], jv = pj[v];
        #pragma unroll
        for (int off = 1; off < 16; off <<= 1) {
            r  += __shfl_xor(r,  off, 32);
            jv += __shfl_xor(jv, off, 32);
        }
        if (m == 0) {
            const int mr = v + 8 * kh;
            const int n  = row0 + mr;
            resid[(size_t)n * NLAT + l]  = r + bout[l];
            wslog[(size_t)l * NTOT + n]  = logf(fabsf(jv));
        }
    }
}

// Deterministic reduction over latents: logabsdet[n] = sum_l log|dJ[l,n]|
__global__ void np_prior_reduce(const float* __restrict__ wslog, float* __restrict__ outlad)
{
    const int n = blockIdx.x * blockDim.x + threadIdx.x;
    if (n < NTOT) {
        float s = 0.f;
        #pragma unroll 8
        for (int l = 0; l < NLAT; ++l) s += wslog[(size_t)l * NTOT + n];
        outlad[n] = s;
    }
}

extern "C" void kernel_launch(void* const* d_in, const int* in_sizes, int n_in,
                              void* d_out, int out_size, void* d_ws, size_t ws_size,
                              hipStream_t stream) {
    const float* x    = (const float*)d_in[0];
    const float* W0   = (const float*)d_in[1];
    const float* b0   = (const float*)d_in[2];
    const float* W1   = (const float*)d_in[3];
    const float* b1   = (const float*)d_in[4];
    const float* W2   = (const float*)d_in[5];
    const float* b2   = (const float*)d_in[6];
    const float* Wout = (const float*)d_in[7];
    const float* bout = (const float*)d_in[8];
    float* out = (float*)d_out;
    float* ws  = (float*)d_ws;  // needs 64*4096*4 = 1 MB

    dim3 grid(64, 64);   // 64 row-blocks (64 rows each) x 64 latents
    dim3 blk(128);       // 4 waves
    np_prior_main<<<grid, blk, 0, stream>>>(x, W0, b0, W1, b1, W2, b2, Wout, bout, out, ws);
    np_prior_reduce<<<(NTOT + 255) / 256, 256, 0, stream>>>(ws, out + RESID_ELEMS);
}